// RoIAlign_35184372089304
// MI455X (gfx1250) — compile-verified
//
#include <hip/hip_runtime.h>

// RoIAlign for MI455X (gfx1250), TDM-staged.
// features: (4,256,200,200) f32, rois: (1000,5) f32 -> out: (1000,256,7,7) f32.
// Strategy: per (roi, 16-channel chunk) workgroup, DMA the RoI's feature
// sub-box (<=34x34 px x 16 ch, 74KB) into LDS with one tensor_load_to_lds
// (3D descriptor: x=tile_w, y=tile_h rows stride W, z=16 channels stride H*W),
// then do all bilinear gathers from LDS. Turns ~800MB of scattered 4B global
// gathers into ~330MB of contiguous DMA + LDS reads.

#define R_N   1000
#define C_N   256
#define H_N   200
#define W_N   200
#define PH    7
#define PW    7
#define NBIN  (PH * PW)
#define SCALE 0.0625f
#define CH    16           // channels staged per block
#define TILE_MAX 34        // max tile extent (roi <= 30 feat px + bilinear halo)

typedef __attribute__((ext_vector_type(4))) unsigned int v4u;
typedef __attribute__((ext_vector_type(8))) int          v8i;
typedef __attribute__((ext_vector_type(4))) int          v4i;

__global__ __launch_bounds__(256)
void roi_align_tdm_kernel(const float* __restrict__ feat,
                          const float* __restrict__ rois,
                          float* __restrict__ out) {
    __shared__ float tile[CH * TILE_MAX * TILE_MAX];   // ~74 KB

    const int r   = blockIdx.y;
    const int c0  = blockIdx.x * CH;
    const int tid = threadIdx.x;

    // --- per-RoI parameters (uniform across the block) ---
    const float bf  = rois[r * 5 + 0];
    const float x1s = rois[r * 5 + 1] * SCALE;
    const float y1s = rois[r * 5 + 2] * SCALE;
    const float x2s = rois[r * 5 + 3] * SCALE;
    const float y2s = rois[r * 5 + 4] * SCALE;
    const float roi_w = fmaxf(x2s - x1s, 1.0f);
    const float roi_h = fmaxf(y2s - y1s, 1.0f);
    const float bin_w = roi_w * (1.0f / PW);
    const float bin_h = roi_h * (1.0f / PH);

    // Tile bounds: cover every clamped bilinear corner [x0 .. x0+tw-1].
    int x0 = (int)floorf(x1s); x0 = x0 < 0 ? 0 : (x0 > W_N - 2 ? W_N - 2 : x0);
    int y0 = (int)floorf(y1s); y0 = y0 < 0 ? 0 : (y0 > H_N - 2 ? H_N - 2 : y0);
    int xE = (int)floorf(x1s + roi_w) + 1; if (xE > W_N - 1) xE = W_N - 1;
    int yE = (int)floorf(y1s + roi_h) + 1; if (yE > H_N - 1) yE = H_N - 1;
    int tw = xE - x0 + 1; if (tw < 2) tw = 2; if (tw > TILE_MAX) tw = TILE_MAX;
    int th = yE - y0 + 1; if (th < 2) th = 2; if (th > TILE_MAX) th = TILE_MAX;

    int b  = (int)bf;
    x0 = __builtin_amdgcn_readfirstlane(x0);
    y0 = __builtin_amdgcn_readfirstlane(y0);
    tw = __builtin_amdgcn_readfirstlane(tw);
    th = __builtin_amdgcn_readfirstlane(th);
    b  = __builtin_amdgcn_readfirstlane(b);

    // --- TDM: stage (tw x th x CH) sub-box global -> LDS, one wavefront lane ---
    if (tid == 0) {
        unsigned long long ga = (unsigned long long)(const void*)
            (feat + ((size_t)b * C_N + (size_t)c0) * (size_t)(H_N * W_N)
                  + (size_t)y0 * W_N + (size_t)x0);
        unsigned lds_off = (unsigned)(size_t)(void*)&tile[0];

        // D# group 0: count=1 | lds_addr | global_addr[56:0] | type=2
        v4u g0;
        g0[0] = 1u;
        g0[1] = lds_off;
        g0[2] = (unsigned)ga;
        g0[3] = (unsigned)((ga >> 32) & 0x01FFFFFFu) | (2u << 30);

        // D# group 1: data_size=4B; tensor_dim0/1 = tile dims (tile built
        // in-bounds so OOB never fires); tile dims; strides W, H*W (elements).
        v8i g1;
        g1[0] = (int)(2u << 16);                                   // data_size=2 (4B)
        g1[1] = (int)((unsigned)tw << 16);                         // tensor_dim0 lo
        g1[2] = (int)((unsigned)th << 16);                         // td0 hi=0 | td1 lo
        g1[3] = (int)((unsigned)tw << 16);                         // td1 hi=0 | tile_dim0
        g1[4] = (int)(((unsigned)CH << 16) | (unsigned)th);        // tile_dim1 | tile_dim2
        g1[5] = W_N;                                               // dim0_stride lo
        g1[6] = (int)(((unsigned)(H_N * W_N) & 0xFFFFu) << 16);    // s0 hi=0 | s1 lo
        g1[7] = (int)((unsigned)(H_N * W_N) >> 16);                // s1 hi

        v4i g2; g2[0] = CH; g2[1] = 0; g2[2] = 0; g2[3] = 0;       // tensor_dim2
        v4i g3; g3[0] = 0;  g3[1] = 0; g3[2] = 0; g3[3] = 0;
        v8i g4; g4[0] = 0;  g4[1] = 0; g4[2] = 0; g4[3] = 0;       // extra group
        g4[4] = 0; g4[5] = 0; g4[6] = 0; g4[7] = 0;                // (6-arg form), zeroed

        __builtin_amdgcn_tensor_load_to_lds(g0, g1, g2, g3, g4, 0);
        __builtin_amdgcn_s_wait_tensorcnt(0);
    }
    __syncthreads();

    // --- bilinear pooling out of LDS ---
    const float fH = (float)H_N, fW = (float)W_N;
    const int plane = th * tw;

    for (int e = tid; e < NBIN * CH; e += 256) {
        const int bin = e % NBIN;
        const int c   = e / NBIN;
        const int ph  = bin / PW;
        const int pw  = bin % PW;
        const float* tp = &tile[c * plane];
        float acc = 0.0f;
        #pragma unroll
        for (int sy = 0; sy < 2; ++sy) {
            const float yy = y1s + ((float)ph + 0.25f + 0.5f * (float)sy) * bin_h;
            const bool vy = (yy >= -1.0f) && (yy <= fH);
            float ycl = fminf(fmaxf(yy, 0.0f), fH - 1.0f);
            float ylf = floorf(ycl);
            if (ylf > fH - 2.0f) ylf = fH - 2.0f;     // fold yh=min(yl+1,H-1) case
            const float ly = ycl - ylf;
            int iy = (int)ylf - y0;
            iy = iy < 0 ? 0 : (iy > th - 2 ? th - 2 : iy);
            #pragma unroll
            for (int sx = 0; sx < 2; ++sx) {
                const float xx = x1s + ((float)pw + 0.25f + 0.5f * (float)sx) * bin_w;
                const bool vx = (xx >= -1.0f) && (xx <= fW);
                float xcl = fminf(fmaxf(xx, 0.0f), fW - 1.0f);
                float xlf = floorf(xcl);
                if (xlf > fW - 2.0f) xlf = fW - 2.0f;
                const float lx = xcl - xlf;
                int ix = (int)xlf - x0;
                ix = ix < 0 ? 0 : (ix > tw - 2 ? tw - 2 : ix);

                const float* p = tp + iy * tw + ix;
                const float v00 = p[0],  v01 = p[1];
                const float v10 = p[tw], v11 = p[tw + 1];
                const float hy = 1.0f - ly, hx = 1.0f - lx;
                const float s  = hy * (hx * v00 + lx * v01)
                               + ly * (hx * v10 + lx * v11);
                acc += (vy && vx) ? s : 0.0f;
            }
        }
        out[((size_t)r * C_N + (size_t)(c0 + c)) * NBIN + bin] = acc * 0.25f;
    }
}

extern "C" void kernel_launch(void* const* d_in, const int* in_sizes, int n_in,
                              void* d_out, int out_size, void* d_ws, size_t ws_size,
                              hipStream_t stream) {
    (void)in_sizes; (void)n_in; (void)out_size; (void)d_ws; (void)ws_size;
    const float* feat = (const float*)d_in[0];
    const float* rois = (const float*)d_in[1];
    float* out = (float*)d_out;
    dim3 grid(C_N / CH, R_N);   // (16 channel-chunks, 1000 rois)
    roi_align_tdm_kernel<<<grid, 256, 0, stream>>>(feat, rois, out);
}